// SigLoss_65266323030280
// MI455X (gfx1250) — compile-verified
//
#include <hip/hip_runtime.h>

typedef __attribute__((ext_vector_type(2))) float v2f;
typedef __attribute__((ext_vector_type(8))) float v8f;

#define BATCH 128
#define PLEN  512   // path length L
#define DIM   16    // feature dim D
#define NROW  511   // L-1 rows of the PDE grid
#define NCOL  512   // row length M+1
#define INCP  (NCOL + 4)  // padded inc row stride (avoid LDS bank conflicts)

// Inclusive scan across a wave32 via lane shuffles (no LDS, no barriers).
__device__ __forceinline__ float wave_incl_scan(float v, int lane) {
#pragma unroll
  for (int off = 1; off < 32; off <<= 1) {
    float n = __shfl_up(v, off, 32);
    if (lane >= off) v += n;
  }
  return v;
}

// Init kernel: overwrite poisoned d_out with the initial-point MSE term.
// mean over all (b,d) of (X[b,0,d] - Y[b,0,d])^2
__global__ __launch_bounds__(256) void sig_init_kernel(
    const float* __restrict__ X, const float* __restrict__ Y,
    float* __restrict__ out) {
  __shared__ float red[256];
  int tid = threadIdx.x;
  float s = 0.0f;
  for (int i = tid; i < BATCH * DIM; i += 256) {
    int b = i >> 4, d = i & 15;
    float diff = X[(size_t)b * PLEN * DIM + d] - Y[(size_t)b * PLEN * DIM + d];
    s += diff * diff;
  }
  red[tid] = s;
  __syncthreads();
#pragma unroll
  for (int off = 128; off > 0; off >>= 1) {
    if (tid < off) red[tid] += red[tid + off];
    __syncthreads();
  }
  if (tid == 0) out[0] = red[0] * (1.0f / (BATCH * DIM));
}

// Main kernel: one workgroup (512 threads = 16 waves) per (pair, batch) problem.
// pair p: 0 -> k(x,x), 1 -> k(y,y), 2 -> k(x,y).
// Per 16-row block: build inc[16][511] tiles with V_WMMA_F32_16X16X4_F32,
// then run the Goursat recurrence (pointwise + workgroup cumsum) row by row.
__global__ __launch_bounds__(512) void sig_goursat_kernel(
    const float* __restrict__ X, const float* __restrict__ Y,
    float* __restrict__ out) {
  __shared__ float inc[16][INCP];      // ~33 KB: one 16-row block of the Gram matrix
  __shared__ float rowbuf[2][NCOL];    // ping-pong PDE rows
  __shared__ float wavesum[16];
  __shared__ float waveoff[16];

  const int tid  = threadIdx.x;
  const int lane = tid & 31;
  const int wid  = tid >> 5;
  const int lm   = lane & 15;   // tile row (A) / tile col (B,D) within 16
  const int lh   = lane >> 4;   // half-wave selector

  const int p = blockIdx.x >> 7;      // 0..2
  const int b = blockIdx.x & 127;     // 0..127
  const float* u = (p == 1) ? Y : X;
  const float* v = (p == 0) ? X : Y;
  u += (size_t)b * PLEN * DIM;
  v += (size_t)b * PLEN * DIM;

  // Boundary row K[0,:] = 1
  rowbuf[0][tid] = 1.0f;
  int cur = 0;
  __syncthreads();

  for (int ib = 0; ib < 32; ++ib) {
    const int row0 = ib * 16;

    // ---- Gram block inc[row0..row0+15][0..511] via fp32 WMMA ----
    // Wave `wid` computes column tiles jt = wid and wid+16 (uniform per wave).
#pragma unroll
    for (int t = 0; t < 2; ++t) {
      const int jt   = wid + t * 16;
      const int col0 = jt * 16;
      v8f acc = {};
#pragma unroll
      for (int kc = 0; kc < 4; ++kc) {
        const int ka = 4 * kc + 2 * lh;  // per-lane K base within chunk
        // A fragment: du[row0+lm][ka], du[row0+lm][ka+1]
        int   ar    = row0 + lm;
        float amask = (ar < NROW) ? 1.0f : 0.0f;
        int   ai    = (ar < NROW) ? ar : (NROW - 1);
        const float* up = u + (size_t)ai * DIM + ka;
        v2f a;
        a.x = amask * (up[DIM + 0] - up[0]);
        a.y = amask * (up[DIM + 1] - up[1]);
        // B fragment: dv[col0+lm][ka], dv[col0+lm][ka+1]  (B[k][n] = dv[n][k])
        int   br    = col0 + lm;
        float bmask = (br < NROW) ? 1.0f : 0.0f;
        int   bi    = (br < NROW) ? br : (NROW - 1);
        const float* vp = v + (size_t)bi * DIM + ka;
        v2f bb;
        bb.x = bmask * (vp[DIM + 0] - vp[0]);
        bb.y = bmask * (vp[DIM + 1] - vp[1]);
        // D = A(16x4) * B(4x16) + C, fp32 accumulate
        acc = __builtin_amdgcn_wmma_f32_16x16x4_f32(
            false, a, false, bb, (short)0, acc, false, false);
      }
      // D layout: VGPR r0 -> row r0 (lanes 0-15) / row r0+8 (lanes 16-31)
#pragma unroll
      for (int r0 = 0; r0 < 8; ++r0) {
        inc[r0 + 8 * lh][col0 + lm] = acc[r0];
      }
    }
    __syncthreads();

    // ---- Goursat recurrence over the 16 rows of this block ----
    const int rmax = (NROW - row0 < 16) ? (NROW - row0) : 16;
    for (int r = 0; r < rmax; ++r) {
      const float* Kp = rowbuf[cur];
      float*       Kn = rowbuf[cur ^ 1];
      float c = 0.0f;
      if (tid < NROW) {  // j = tid in 0..510
        c = Kp[tid + 1] + Kp[tid] * (inc[r][tid] - 1.0f);
      }
      float s = wave_incl_scan(c, lane);
      if (lane == 31) wavesum[wid] = s;
      __syncthreads();
      if (wid == 0) {
        float wv = (lane < 16) ? wavesum[lane] : 0.0f;
        float ws = wave_incl_scan(wv, lane);
        if (lane < 16) waveoff[lane] = ws - wv;  // exclusive offsets
      }
      __syncthreads();
      float total = s + waveoff[wid];  // full inclusive cumsum of c
      if (tid == 0)    Kn[0] = 1.0f;
      if (tid < NROW)  Kn[tid + 1] = 1.0f + total;
      __syncthreads();   // also guards inc[] reuse by next block's WMMA stores
      cur ^= 1;
    }
  }

  if (tid == 0) {
    float kval = rowbuf[cur][NCOL - 1];       // K[L-1, M-1]
    float sign = (p == 2) ? -2.0f : 1.0f;     // mean(kxx)+mean(kyy)-2*mean(kxy)
    atomicAdd(out, sign * kval * (1.0f / BATCH));
  }
}

extern "C" void kernel_launch(void* const* d_in, const int* in_sizes, int n_in,
                              void* d_out, int out_size, void* d_ws, size_t ws_size,
                              hipStream_t stream) {
  (void)in_sizes; (void)n_in; (void)out_size; (void)d_ws; (void)ws_size;
  const float* X = (const float*)d_in[0];
  const float* Y = (const float*)d_in[1];
  float* out = (float*)d_out;

  // Order on one stream: init writes out[0] (MSE term), main kernel accumulates.
  sig_init_kernel<<<1, 256, 0, stream>>>(X, Y, out);
  sig_goursat_kernel<<<3 * BATCH, 512, 0, stream>>>(X, Y, out);
}